// Max_AVG_Pooling_Layer_71966472011821
// MI455X (gfx1250) — compile-verified
//
#include <hip/hip_runtime.h>

typedef __attribute__((ext_vector_type(2))) float v2f;
typedef __attribute__((ext_vector_type(8))) float v8f;

#define G    512
#define NPTS 400
#define DIM  128
#define KCL  200
#define DOUTN 512
#define DMID 32
#define GK   (G * KCL)     // 102400 pooled rows
#define KD   (KCL * DIM)   // 25600
#define KS   50            // split-K slices for GEMM2
#define KSL  (KD / KS)     // 512 k per slice

static __device__ __forceinline__ void atomAddF(float* p, float v) {
    __hip_atomic_fetch_add(p, v, __ATOMIC_RELAXED, __HIP_MEMORY_SCOPE_AGENT);
}

// ---- zero scratch (sums | cnt | h2acc are contiguous at ws start) + d_out ----
__global__ void k_zero(float* __restrict__ ws, long n, float* __restrict__ out) {
    long i = (long)blockIdx.x * blockDim.x + threadIdx.x;
    if (i < n) ws[i] = 0.0f;
    if (i == 0) out[0] = 0.0f;
}

// ---- segment sum: one block (128 threads) per input row ----
__global__ void k_segsum(const float* __restrict__ x, const int* __restrict__ cl,
                         float* __restrict__ sums, float* __restrict__ cnt) {
    int row = blockIdx.x;              // 0 .. G*NPTS-1
    int g   = row / NPTS;
    int c   = cl[row];
    long base = ((long)g * KCL + c) * DIM;
    atomAddF(&sums[base + threadIdx.x], x[(long)row * DIM + threadIdx.x]);
    if (threadIdx.x == 0) atomAddF(&cnt[g * KCL + c], 1.0f);
}

// ---- GEMM1: h1 = relu(avg(pooled) @ W1 + b1), fp32 WMMA 16x16x4 ----
// block = 256 threads = 8 waves; wave w owns ntile w; blockIdx.x = mtile
__global__ void k_gemm1(const float* __restrict__ sums, const float* __restrict__ cnt,
                        const float* __restrict__ W1, const float* __restrict__ b1,
                        float* __restrict__ h1) {
    int mtile  = blockIdx.x;           // 0..GK/16-1
    int wave   = threadIdx.x >> 5;     // 0..7
    int lane   = threadIdx.x & 31;
    int laneLo = lane & 15;
    int hi     = lane >> 4;            // 0: K 0/1, 1: K 2/3
    int n0     = wave * 16;

    int arow = mtile * 16 + laneLo;
    float invc = 1.0f / fmaxf(cnt[arow], 1.0f);
    const float* arp = sums + (long)arow * DIM;

    v8f c = {};
    for (int k = 0; k < DIM; k += 4) {
        int ka = k + hi * 2;
        v2f a = *(const v2f*)(arp + ka);   // adjacent K -> 8B load
        a.x *= invc; a.y *= invc;
        v2f b;
        b.x = W1[(long)ka * DIM + n0 + laneLo];
        b.y = W1[(long)(ka + 1) * DIM + n0 + laneLo];
        c = __builtin_amdgcn_wmma_f32_16x16x4_f32(false, a, false, b,
                                                  (short)0, c, false, false);
    }
    int orow = mtile * 16 + hi * 8;
    int ocol = n0 + laneLo;
    float bias = b1[ocol];
#pragma unroll
    for (int j = 0; j < 8; ++j) {
        float v = c[j] + bias;
        h1[(long)(orow + j) * DIM + ocol] = fmaxf(v, 0.0f);
    }
}

// ---- GEMM2: (512 x 25600) @ (25600 x 32), split-K WMMA, atomics into h2acc ----
// one wave (32 threads) per block; blockIdx.x encodes (mtile, ntile, kslice)
__global__ void k_gemm2(const float* __restrict__ h1, const float* __restrict__ W2,
                        float* __restrict__ h2acc) {
    int bid    = blockIdx.x;
    int mtile  = bid & 31;             // 32 mtiles
    int nt     = (bid >> 5) & 1;       // 2 ntiles
    int ks     = bid >> 6;             // KS slices
    int lane   = threadIdx.x & 31;
    int laneLo = lane & 15;
    int hi     = lane >> 4;
    int n0     = nt * 16;

    const float* arp = h1 + (long)(mtile * 16 + laneLo) * KD;   // h1 viewed (512, 25600)
    const float* brp = W2 + (long)(n0 + laneLo) * KD;           // W2 is (32, 25600), B=W2^T

    v8f c = {};
    int k0 = ks * KSL;
    for (int k = k0; k < k0 + KSL; k += 4) {
        int ka = k + hi * 2;
        v2f a = *(const v2f*)(arp + ka);
        v2f b = *(const v2f*)(brp + ka);
        c = __builtin_amdgcn_wmma_f32_16x16x4_f32(false, a, false, b,
                                                  (short)0, c, false, false);
    }
    int orow = mtile * 16 + hi * 8;
    int ocol = n0 + laneLo;
#pragma unroll
    for (int j = 0; j < 8; ++j)
        atomAddF(&h2acc[(orow + j) * DMID + ocol], c[j]);
}

// ---- bias + relu after split-K reduction ----
__global__ void k_bias2(const float* __restrict__ acc, const float* __restrict__ b2,
                        float* __restrict__ h2) {
    int i = blockIdx.x * blockDim.x + threadIdx.x;
    if (i < G * DMID) h2[i] = fmaxf(acc[i] + b2[i % DMID], 0.0f);
}

// ---- GEMM3: (512 x 32) @ (32 x 512) + b3, relu (tiny, VALU) ----
__global__ void k_gemm3(const float* __restrict__ h2, const float* __restrict__ W3,
                        const float* __restrict__ b3, float* __restrict__ h3) {
    int i = blockIdx.x * blockDim.x + threadIdx.x;   // G*DOUTN threads
    int m = i / DOUTN, n = i % DOUTN;
    float s = b3[n];
    const float* hr = h2 + m * DMID;
    const float* wr = W3 + n * DMID;
#pragma unroll
    for (int j = 0; j < DMID; ++j) s += hr[j] * wr[j];
    h3[i] = fmaxf(s, 0.0f);
}

// ---- cosine similarity pairs + MSE (one wave per pair) ----
__global__ void k_loss(const float* __restrict__ h3, const float* __restrict__ ts,
                       float* __restrict__ out) {
    int wave = threadIdx.x >> 5;
    int lane = threadIdx.x & 31;
    int pair = blockIdx.x * 8 + wave;          // 0..255
    const float* g1 = h3 + (long)(2 * pair) * DOUTN;
    const float* g2 = g1 + DOUTN;
    float dot = 0.0f, s1 = 0.0f, s2 = 0.0f;
    for (int i = lane; i < DOUTN; i += 32) {
        float a = g1[i], b = g2[i];
        dot += a * b; s1 += a * a; s2 += b * b;
    }
#pragma unroll
    for (int off = 16; off >= 1; off >>= 1) {
        dot += __shfl_xor(dot, off, 32);
        s1  += __shfl_xor(s1,  off, 32);
        s2  += __shfl_xor(s2,  off, 32);
    }
    if (lane == 0) {
        const float eps = 1e-6f;
        float score = dot / (fmaxf(sqrtf(s1), eps) * fmaxf(sqrtf(s2), eps));
        float d = score - ts[pair];
        atomAddF(out, d * d * (1.0f / 256.0f));
    }
}

extern "C" void kernel_launch(void* const* d_in, const int* in_sizes, int n_in,
                              void* d_out, int out_size, void* d_ws, size_t ws_size,
                              hipStream_t stream) {
    const float* x  = (const float*)d_in[0];
    const int*   cl = (const int*)d_in[1];
    const float* ts = (const float*)d_in[2];
    const float* W1 = (const float*)d_in[3];
    const float* b1 = (const float*)d_in[4];
    const float* W2 = (const float*)d_in[5];
    const float* b2 = (const float*)d_in[6];
    const float* W3 = (const float*)d_in[7];
    const float* b3 = (const float*)d_in[8];
    float* out = (float*)d_out;

    float* ws    = (float*)d_ws;
    float* sums  = ws;                          // GK*DIM
    float* cnt   = sums  + (size_t)GK * DIM;    // GK
    float* h2acc = cnt   + GK;                  // G*DMID
    float* h1    = h2acc + G * DMID;            // GK*DIM
    float* h2    = h1    + (size_t)GK * DIM;    // G*DMID
    float* h3    = h2    + G * DMID;            // G*DOUTN

    long nz = (long)GK * DIM + GK + G * DMID;   // sums|cnt|h2acc contiguous
    k_zero<<<(unsigned)((nz + 255) / 256), 256, 0, stream>>>(ws, nz, out);
    k_segsum<<<G * NPTS, DIM, 0, stream>>>(x, cl, sums, cnt);
    k_gemm1<<<GK / 16, 256, 0, stream>>>(sums, cnt, W1, b1, h1);
    k_gemm2<<<32 * 2 * KS, 32, 0, stream>>>(h1, W2, h2acc);
    k_bias2<<<(G * DMID + 255) / 256, 256, 0, stream>>>(h2acc, b2, h2);
    k_gemm3<<<(G * DOUTN + 255) / 256, 256, 0, stream>>>(h2, W3, b3, h3);
    k_loss<<<32, 256, 0, stream>>>(h3, ts, out);
}